// BDRRAA_18614388261041
// MI455X (gfx1250) — compile-verified
//
#include <hip/hip_runtime.h>
#include <math.h>

typedef __attribute__((ext_vector_type(2))) float v2f;
typedef __attribute__((ext_vector_type(8))) float v8f;

#define KDIM 32
#define DDIM 32
#define EPSV 1e-6f

// ---------- softmax over k (columns of (k, n) row-major) ----------
__global__ void softmax_cols_k(const float* __restrict__ Z, float* __restrict__ out, int n) {
    int i = blockIdx.x * blockDim.x + threadIdx.x;
    if (i >= n) return;
    float v[KDIM];
    float mx = -INFINITY;
#pragma unroll
    for (int kk = 0; kk < KDIM; ++kk) { v[kk] = Z[kk * n + i]; mx = fmaxf(mx, v[kk]); }
    float s = 0.f;
#pragma unroll
    for (int kk = 0; kk < KDIM; ++kk) { v[kk] = expf(v[kk] - mx); s += v[kk]; }
    float inv = 1.f / s;
#pragma unroll
    for (int kk = 0; kk < KDIM; ++kk) out[kk * n + i] = v[kk] * inv;
}

// ---------- gather sampled columns into Zg (k, m), record raw gate index ----------
__global__ void gather_Z_k(const float* __restrict__ Zi, const float* __restrict__ Zj,
                           const int* __restrict__ si, const int* __restrict__ sj,
                           float* __restrict__ Zg, int* __restrict__ idxArr,
                           int n_i, int n_j, int m_i, int m) {
    int c = blockIdx.x * blockDim.x + threadIdx.x;
    if (c >= m) return;
    int idx; const float* src; int n;
    if (c < m_i) { idx = si[c];        src = Zi; n = n_i; }
    else         { idx = sj[c - m_i];  src = Zj; n = n_j; }  // NOTE: raw index, no +n_i (faithful)
    idxArr[c] = idx;
#pragma unroll
    for (int kk = 0; kk < KDIM; ++kk) Zg[kk * m + c] = src[kk * n + idx];
}

// ---------- S[a,b] = sum_c Z[a,c]*Z[b,c]*sigmoid(Gate[idx_c, b]); colsum[b] ----------
__global__ void compute_S_k(const float* __restrict__ Zg, const int* __restrict__ idxArr,
                            const float* __restrict__ Gate,
                            float* __restrict__ S, float* __restrict__ colsum, int m) {
    int a = blockIdx.x >> 5;
    int b = blockIdx.x & 31;
    __shared__ float r1[256];
    __shared__ float r2[256];
    float acc = 0.f, cs = 0.f;
    for (int c = threadIdx.x; c < m; c += blockDim.x) {
        float va = Zg[a * m + c];
        float vb = Zg[b * m + c];
        float g  = 1.f / (1.f + expf(-Gate[idxArr[c] * KDIM + b]));
        float zg = vb * g;
        acc += va * zg;
        cs  += zg;
    }
    r1[threadIdx.x] = acc; r2[threadIdx.x] = cs;
    __syncthreads();
    for (int off = 128; off > 0; off >>= 1) {
        if ((int)threadIdx.x < off) { r1[threadIdx.x] += r1[threadIdx.x + off];
                                      r2[threadIdx.x] += r2[threadIdx.x + off]; }
        __syncthreads();
    }
    if (threadIdx.x == 0) {
        S[a * KDIM + b] = r1[0];
        if (a == 0) colsum[b] = r2[0];
    }
}

// ---------- AZC = A @ (S / colsum) — single block, 1024 threads ----------
__global__ void finalize_AZC_k(const float* __restrict__ S, const float* __restrict__ colsum,
                               const float* __restrict__ A, float* __restrict__ AZC) {
    __shared__ float M[KDIM * KDIM];
    int t = threadIdx.x;          // 0..1023
    int a = t >> 5, b = t & 31;
    M[t] = S[t] / colsum[b];
    __syncthreads();
    float acc = 0.f;
#pragma unroll
    for (int kk = 0; kk < KDIM; ++kk) acc += A[a * KDIM + kk] * M[kk * KDIM + b];
    AZC[t] = acc;                 // (d=a, k=b)
}

// ---------- latent positions of sampled nodes (padded), norms, biases ----------
// X[t,:] = AZC @ Zsm[:, idx] - eps   (eps=0 for i side, EPS for j side -> yj)
__global__ void gather_x_k(const float* __restrict__ AZC, const float* __restrict__ Zsm,
                           const int* __restrict__ sidx, const float* __restrict__ biasvec,
                           float* __restrict__ X, float* __restrict__ nrm, float* __restrict__ bs,
                           int n, int m, int pad, float eps) {
    __shared__ float Al[DDIM * KDIM];
    for (int t = threadIdx.x; t < DDIM * KDIM; t += blockDim.x) Al[t] = AZC[t];
    __syncthreads();
    int t = blockIdx.x * blockDim.x + threadIdx.x;
    if (t >= pad) return;
    if (t < m) {
        int idx = sidx[t];
        float z[KDIM];
#pragma unroll
        for (int kk = 0; kk < KDIM; ++kk) z[kk] = Zsm[kk * n + idx];
        float nn = 0.f;
#pragma unroll 4
        for (int dd = 0; dd < DDIM; ++dd) {
            float x = 0.f;
#pragma unroll
            for (int kk = 0; kk < KDIM; ++kk) x += Al[dd * KDIM + kk] * z[kk];
            x -= eps;
            X[t * DDIM + dd] = x;
            nn += x * x;
        }
        nrm[t] = nn;
        bs[t]  = biasvec[idx];
    } else {
        for (int dd = 0; dd < DDIM; ++dd) X[t * DDIM + dd] = 0.f;
        nrm[t] = 0.f;
        bs[t]  = -1e30f;   // exp(bias - dist) -> 0 for pad rows/cols
    }
}

// ---------- pairwise term: Gram via v_wmma_f32_16x16x4_f32, fused exp-sum ----------
__global__ void pair_kernel(const float* __restrict__ Xi, const float* __restrict__ Yj,
                            const float* __restrict__ ni, const float* __restrict__ nj,
                            const float* __restrict__ bi, const float* __restrict__ bj,
                            float* __restrict__ partial, int TI, int TJ) {
    int wave = threadIdx.x >> 5;
    int lane = threadIdx.x & 31;
    int tileId = blockIdx.x * 8 + wave;
    float sum = 0.f;
    if (tileId < TI * TJ) {
        int tr = tileId / TJ, tc = tileId % TJ;
        int half = lane >> 4, lidx = lane & 15;
        int row16 = tr * 16 + lidx;   // A: lanes 0-15 & 16-31 both hold M=0..15
        int col16 = tc * 16 + lidx;   // B: mirrored layout over N
        v8f acc = {};
#pragma unroll
        for (int k0 = 0; k0 < KDIM; k0 += 4) {
            int ko = k0 + 2 * half;   // lane<16: K={k0,k0+1}; lane>=16: K={k0+2,k0+3}
            v2f a = *(const v2f*)(Xi + row16 * KDIM + ko);
            v2f b = *(const v2f*)(Yj + col16 * KDIM + ko);
            acc = __builtin_amdgcn_wmma_f32_16x16x4_f32(
                false, a, false, b, (short)0, acc, false, false);
        }
        // D layout: VGPR r, lanes 0-15 -> (M=r, N=lidx); lanes 16-31 -> (M=r+8, N=lidx)
        float njv = nj[col16];
        float gj  = bj[col16];
        int rowbase = tr * 16 + 8 * half;
#pragma unroll
        for (int r = 0; r < 8; ++r) {
            int row = rowbase + r;
            float d2 = ni[row] - 2.f * acc[r] + njv;   // ||xi - (xj-EPS)||^2
            d2 = fmaxf(d2, 0.f);
            float t = bi[row] + gj - sqrtf(d2);
            sum += expf(t);
        }
    }
    __shared__ float red[256];
    red[threadIdx.x] = sum;
    __syncthreads();
    for (int off = 128; off > 0; off >>= 1) {
        if ((int)threadIdx.x < off) red[threadIdx.x] += red[threadIdx.x + off];
        __syncthreads();
    }
    if (threadIdx.x == 0) partial[blockIdx.x] = red[0];
}

// ---------- edge term: v = AZC @ (zi - zj); term = beta+gamma - ||v+EPS|| ----------
__global__ void edge_kernel(const float* __restrict__ AZC,
                            const float* __restrict__ Zi, const float* __restrict__ Zj,
                            const int* __restrict__ ei, const int* __restrict__ ej,
                            const float* __restrict__ beta, const float* __restrict__ gamma,
                            float* __restrict__ partial, int n_i, int n_j, int E) {
    __shared__ float Al[DDIM * KDIM];
    for (int t = threadIdx.x; t < DDIM * KDIM; t += blockDim.x) Al[t] = AZC[t];
    __syncthreads();
    int e = blockIdx.x * blockDim.x + threadIdx.x;
    float term = 0.f;
    if (e < E) {
        int ii = ei[e], jj = ej[e];
        float w[KDIM];
#pragma unroll
        for (int kk = 0; kk < KDIM; ++kk) w[kk] = Zi[kk * n_i + ii] - Zj[kk * n_j + jj];
        float s2 = 0.f;
#pragma unroll 4
        for (int dd = 0; dd < DDIM; ++dd) {
            float v = 0.f;
#pragma unroll
            for (int kk = 0; kk < KDIM; ++kk) v += Al[dd * KDIM + kk] * w[kk];
            v += EPSV;
            s2 += v * v;
        }
        term = beta[ii] + gamma[jj] - sqrtf(s2);
    }
    __shared__ float red[256];
    red[threadIdx.x] = term;
    __syncthreads();
    for (int off = 128; off > 0; off >>= 1) {
        if ((int)threadIdx.x < off) red[threadIdx.x] += red[threadIdx.x + off];
        __syncthreads();
    }
    if (threadIdx.x == 0) partial[blockIdx.x] = red[0];
}

// ---------- final: out = sum(edge partials) - sum(pair partials) ----------
__global__ void final_reduce_k(const float* __restrict__ edgeP, int nE,
                               const float* __restrict__ pairP, int nP,
                               float* __restrict__ out) {
    __shared__ float rl[256], rm[256];
    float l = 0.f, p = 0.f;
    for (int i = threadIdx.x; i < nE; i += blockDim.x) l += edgeP[i];
    for (int i = threadIdx.x; i < nP; i += blockDim.x) p += pairP[i];
    rl[threadIdx.x] = l; rm[threadIdx.x] = p;
    __syncthreads();
    for (int off = 128; off > 0; off >>= 1) {
        if ((int)threadIdx.x < off) { rl[threadIdx.x] += rl[threadIdx.x + off];
                                      rm[threadIdx.x] += rm[threadIdx.x + off]; }
        __syncthreads();
    }
    if (threadIdx.x == 0) out[0] = rl[0] - rm[0];
}

extern "C" void kernel_launch(void* const* d_in, const int* in_sizes, int n_in,
                              void* d_out, int out_size, void* d_ws, size_t ws_size,
                              hipStream_t stream) {
    const float* beta  = (const float*)d_in[0];
    const float* gamma = (const float*)d_in[1];
    const float* A     = (const float*)d_in[2];
    const float* Z_i   = (const float*)d_in[3];
    const float* Z_j   = (const float*)d_in[4];
    const float* Gate  = (const float*)d_in[5];
    const int*   si    = (const int*)d_in[6];
    const int*   sj    = (const int*)d_in[7];
    const int*   esi   = (const int*)d_in[8];
    const int*   esj   = (const int*)d_in[9];

    const int n_i = in_sizes[0];
    const int n_j = in_sizes[1];
    const int m_i = in_sizes[6];
    const int m_j = in_sizes[7];
    const int E   = in_sizes[8];
    const int m   = m_i + m_j;
    const int PADI = ((m_i + 15) / 16) * 16;
    const int PADJ = ((m_j + 15) / 16) * 16;
    const int TI = PADI / 16, TJ = PADJ / 16;
    const int nPairBlocks = (TI * TJ + 7) / 8;
    const int nEdgeBlocks = (E + 255) / 256;

    float* ws = (float*)d_ws;
    size_t off = 0;
    float* Zi     = ws + off; off += (size_t)KDIM * n_i;
    float* Zj     = ws + off; off += (size_t)KDIM * n_j;
    float* Zg     = ws + off; off += (size_t)KDIM * m;
    int*   idxArr = (int*)(ws + off); off += (size_t)m;
    float* S      = ws + off; off += KDIM * KDIM;
    float* colsum = ws + off; off += KDIM;
    float* AZC    = ws + off; off += DDIM * KDIM;
    float* Xi     = ws + off; off += (size_t)PADI * DDIM;   // even float offset -> 8B aligned
    float* Yj     = ws + off; off += (size_t)PADJ * DDIM;
    float* ni     = ws + off; off += (size_t)PADI;
    float* njv    = ws + off; off += (size_t)PADJ;
    float* bi     = ws + off; off += (size_t)PADI;
    float* bj     = ws + off; off += (size_t)PADJ;
    float* pairP  = ws + off; off += (size_t)nPairBlocks;
    float* edgeP  = ws + off; off += (size_t)nEdgeBlocks;
    (void)ws_size; (void)n_in; (void)out_size;   // ~2.7 MB total workspace

    softmax_cols_k<<<(n_i + 255) / 256, 256, 0, stream>>>(Z_i, Zi, n_i);
    softmax_cols_k<<<(n_j + 255) / 256, 256, 0, stream>>>(Z_j, Zj, n_j);
    gather_Z_k<<<(m + 255) / 256, 256, 0, stream>>>(Zi, Zj, si, sj, Zg, idxArr, n_i, n_j, m_i, m);
    compute_S_k<<<KDIM * KDIM, 256, 0, stream>>>(Zg, idxArr, Gate, S, colsum, m);
    finalize_AZC_k<<<1, 1024, 0, stream>>>(S, colsum, A, AZC);
    gather_x_k<<<(PADI + 255) / 256, 256, 0, stream>>>(AZC, Zi, si, beta,  Xi, ni,  bi, n_i, m_i, PADI, 0.0f);
    gather_x_k<<<(PADJ + 255) / 256, 256, 0, stream>>>(AZC, Zj, sj, gamma, Yj, njv, bj, n_j, m_j, PADJ, EPSV);
    pair_kernel<<<nPairBlocks, 256, 0, stream>>>(Xi, Yj, ni, njv, bi, bj, pairP, TI, TJ);
    edge_kernel<<<nEdgeBlocks, 256, 0, stream>>>(AZC, Zi, Zj, esi, esj, beta, gamma, edgeP, n_i, n_j, E);
    final_reduce_k<<<1, 256, 0, stream>>>(edgeP, nEdgeBlocks, pairP, nPairBlocks, (float*)d_out);
}